// MessagePassingGNN_4346506904055
// MI455X (gfx1250) — compile-verified
//
#include <hip/hip_runtime.h>
#include <hip/hip_bf16.h>

typedef __attribute__((ext_vector_type(2))) float v2f;
typedef __attribute__((ext_vector_type(8))) float v8f;

#define HEADS 4
#define CH    64
#define HC    256
#define DOUT  384
#define NEG_SLOPE 0.2f
#define LN_EPS 1e-5f

// ---------------------------------------------------------------------------
// Ordered-int key for float atomic max (monotonic map, handles negatives)
// ---------------------------------------------------------------------------
__device__ __forceinline__ unsigned int fkey(float f) {
    unsigned int u = __float_as_uint(f);
    return (u & 0x80000000u) ? ~u : (u | 0x80000000u);
}
__device__ __forceinline__ float funkey(unsigned int k) {
    unsigned int u = (k & 0x80000000u) ? (k & 0x7FFFFFFFu) : ~k;
    return __uint_as_float(u);
}

// ---------------------------------------------------------------------------
// Zero-init
// ---------------------------------------------------------------------------
__global__ void k_zero_f32(float* __restrict__ p, int n) {
    int i = blockIdx.x * blockDim.x + threadIdx.x;
    if (i < n) p[i] = 0.0f;
}

// ---------------------------------------------------------------------------
// fp32 GEMM via V_WMMA_F32_16X16X4_F32.
// Each wave computes a 16-row x 64-col strip of Out = A[MxK] * B[KxNcol].
// Requires: M % 16 == 0, Ncol % 64 == 0, K % 4 == 0 (all hold here).
// A layout per ISA: lanes 0-15 -> M=lane, K = k..k+1 ; lanes 16-31 -> K = k+2..k+3
// B layout per ISA: lanes 0-15 -> N=lane, K = k..k+1 ; lanes 16-31 -> K = k+2..k+3
// D layout: VGPR r -> row m0 + r + 8*(lane>=16), col = n0 + (lane&15)
// ---------------------------------------------------------------------------
__global__ void k_gemm_wmma(const float* __restrict__ A, const float* __restrict__ B,
                            const float* __restrict__ bias, float* __restrict__ Out,
                            int M, int K, int Ncol, int relu) {
    int wid  = (blockIdx.x * blockDim.x + threadIdx.x) >> 5;
    int lane = threadIdx.x & 31;
    int colWaves = Ncol >> 6;                 // 64 cols per wave
    int rowBlock = wid / colWaves;
    int colBlock = wid - rowBlock * colWaves;
    if (rowBlock * 16 >= M) return;           // wave-uniform exit (EXEC all-ones inside)

    int m0   = rowBlock * 16;
    int n0   = colBlock * 64;
    int half = lane >> 4;                     // 0 or 1
    int l    = lane & 15;

    v8f acc0 = {}, acc1 = {}, acc2 = {}, acc3 = {};
    const float* arow = A + (size_t)(m0 + l) * K + half * 2;

    for (int k = 0; k < K; k += 4) {
        v2f a = *(const v2f*)(arow + k);      // A[m0+l][k+2*half .. +1]
        int kk = k + half * 2;
        const float* bp = B + (size_t)kk * Ncol + n0 + l;
        v2f b0, b1, b2, b3;
        b0.x = bp[0];   b0.y = bp[Ncol + 0];
        b1.x = bp[16];  b1.y = bp[Ncol + 16];
        b2.x = bp[32];  b2.y = bp[Ncol + 32];
        b3.x = bp[48];  b3.y = bp[Ncol + 48];
        acc0 = __builtin_amdgcn_wmma_f32_16x16x4_f32(false, a, false, b0, (short)0, acc0, false, false);
        acc1 = __builtin_amdgcn_wmma_f32_16x16x4_f32(false, a, false, b1, (short)0, acc1, false, false);
        acc2 = __builtin_amdgcn_wmma_f32_16x16x4_f32(false, a, false, b2, (short)0, acc2, false, false);
        acc3 = __builtin_amdgcn_wmma_f32_16x16x4_f32(false, a, false, b3, (short)0, acc3, false, false);
    }

#pragma unroll
    for (int r = 0; r < 8; ++r) {
        int row = m0 + r + half * 8;
        float* op = Out + (size_t)row * Ncol + n0 + l;
        float v0 = acc0[r], v1 = acc1[r], v2 = acc2[r], v3 = acc3[r];
        if (bias) {
            v0 += bias[n0 + l];
            v1 += bias[n0 + 16 + l];
            v2 += bias[n0 + 32 + l];
            v3 += bias[n0 + 48 + l];
        }
        if (relu) {
            v0 = fmaxf(v0, 0.0f); v1 = fmaxf(v1, 0.0f);
            v2 = fmaxf(v2, 0.0f); v3 = fmaxf(v3, 0.0f);
        }
        op[0] = v0; op[16] = v1; op[32] = v2; op[48] = v3;
    }
}

// ---------------------------------------------------------------------------
// a_s[n,h] = dot(hfeat[n, h*64 : h*64+64], att_s[h]); a_d likewise
// ---------------------------------------------------------------------------
__global__ void k_attn_coef(const float* __restrict__ hfeat,
                            const float* __restrict__ as_w, const float* __restrict__ ad_w,
                            float* __restrict__ a_s, float* __restrict__ a_d, int N) {
    int idx = blockIdx.x * blockDim.x + threadIdx.x;   // (n,h)
    if (idx >= N * HEADS) return;
    int n = idx >> 2, h = idx & 3;
    const float* hp = hfeat + (size_t)n * HC + h * CH;
    const float* sp = as_w + h * CH;
    const float* dp = ad_w + h * CH;
    float ss = 0.0f, sd = 0.0f;
#pragma unroll 4
    for (int c = 0; c < CH; ++c) {
        float v = hp[c];
        ss += v * sp[c];
        sd += v * dp[c];
    }
    a_s[idx] = ss;
    a_d[idx] = sd;
}

// ---------------------------------------------------------------------------
// Edge pass A: e = leaky_relu(a_s[src]+a_d[dst]); segment max via ordered-int
// ---------------------------------------------------------------------------
__global__ void k_edge_logits(const int* __restrict__ ei, int E, int Etot,
                              const float* __restrict__ a_s, const float* __restrict__ a_d,
                              float* __restrict__ eatt, unsigned int* __restrict__ mkey) {
    int idx = blockIdx.x * blockDim.x + threadIdx.x;   // (e,h)
    if (idx >= Etot * HEADS) return;
    int e = idx >> 2, h = idx & 3;
    int s, d;
    if (e < E) { s = ei[e]; d = ei[E + e]; } else { s = d = e - E; }
    float v = a_s[s * HEADS + h] + a_d[d * HEADS + h];
    v = v > 0.0f ? v : NEG_SLOPE * v;
    eatt[idx] = v;
    atomicMax(&mkey[d * HEADS + h], fkey(v));
}

// ---------------------------------------------------------------------------
// Edge pass B: e <- exp(e - m[dst]); segment sum
// ---------------------------------------------------------------------------
__global__ void k_edge_exp(const int* __restrict__ ei, int E, int Etot,
                           const unsigned int* __restrict__ mkey,
                           float* __restrict__ eatt, float* __restrict__ ssum) {
    int idx = blockIdx.x * blockDim.x + threadIdx.x;   // (e,h)
    if (idx >= Etot * HEADS) return;
    int e = idx >> 2, h = idx & 3;
    int d;
    if (e < E) { d = ei[E + e]; } else { d = e - E; }
    float m = funkey(mkey[d * HEADS + h]);
    float w = __expf(eatt[idx] - m);
    eatt[idx] = w;
    atomicAdd(&ssum[d * HEADS + h], w);
}

// ---------------------------------------------------------------------------
// Edge pass C: agg[dst] += hfeat[src] * (e / s[dst]), per (head,channel)
// One 256-thread block covers the full 256-wide feature row; loops over edges.
// ---------------------------------------------------------------------------
#define EDGES_PER_BLOCK 4
__global__ void k_edge_aggregate(const int* __restrict__ ei, int E, int Etot,
                                 const float* __restrict__ hfeat,
                                 const float* __restrict__ eatt, const float* __restrict__ ssum,
                                 float* __restrict__ agg) {
    int t = threadIdx.x;          // 0..255 = h*64 + c
    int h = t >> 6;
    int e0 = blockIdx.x * EDGES_PER_BLOCK;
    for (int i = 0; i < EDGES_PER_BLOCK; ++i) {
        int e = e0 + i;
        if (e >= Etot) return;
        int s, d;
        if (e < E) { s = ei[e]; d = ei[E + e]; } else { s = d = e - E; }
        float alpha = eatt[e * HEADS + h] / ssum[d * HEADS + h];
        float val = hfeat[(size_t)s * HC + t] * alpha;
        atomicAdd(&agg[(size_t)d * HC + t], val);
    }
}

// ---------------------------------------------------------------------------
// In-place: buf[n,:] = relu(layernorm(buf[n,:] + bias) * gamma + beta)
// ---------------------------------------------------------------------------
__global__ void k_bias_ln_relu(float* __restrict__ buf, const float* __restrict__ bias,
                               const float* __restrict__ gamma, const float* __restrict__ beta) {
    __shared__ float red[HC];
    int n = blockIdx.x;
    int t = threadIdx.x;
    float v = buf[(size_t)n * HC + t] + bias[t];
    red[t] = v;
    __syncthreads();
    for (int s = HC / 2; s > 0; s >>= 1) {
        if (t < s) red[t] += red[t + s];
        __syncthreads();
    }
    float mu = red[0] * (1.0f / HC);
    __syncthreads();
    float dv = v - mu;
    red[t] = dv * dv;
    __syncthreads();
    for (int s = HC / 2; s > 0; s >>= 1) {
        if (t < s) red[t] += red[t + s];
        __syncthreads();
    }
    float var = red[0] * (1.0f / HC);
    float o = dv * rsqrtf(var + LN_EPS) * gamma[t] + beta[t];
    buf[(size_t)n * HC + t] = fmaxf(o, 0.0f);
}

// ---------------------------------------------------------------------------
// Column sums over all nodes -> acc[256] (atomic per-block partials)
// ---------------------------------------------------------------------------
__global__ void k_colsum(const float* __restrict__ buf, float* __restrict__ acc, int N) {
    int t = threadIdx.x;
    int chunk = (N + gridDim.x - 1) / gridDim.x;
    int n0 = blockIdx.x * chunk;
    int n1 = n0 + chunk; if (n1 > N) n1 = N;
    float s = 0.0f;
    for (int n = n0; n < n1; ++n) s += buf[(size_t)n * HC + t];
    atomicAdd(&acc[t], s);
}

// ---------------------------------------------------------------------------
// out[d] = b_out[d] + sum_k (colsum[k]/N) * w_out[k, d]
// ---------------------------------------------------------------------------
__global__ void k_final_proj(const float* __restrict__ colsum, const float* __restrict__ Wout,
                             const float* __restrict__ bout, float* __restrict__ out, float invN) {
    int d = blockIdx.x * blockDim.x + threadIdx.x;
    if (d >= DOUT) return;
    float s = bout[d];
    for (int k = 0; k < HC; ++k) s += colsum[k] * invN * Wout[(size_t)k * DOUT + d];
    out[d] = s;
}

// ---------------------------------------------------------------------------
// Host-side orchestration
// ---------------------------------------------------------------------------
static inline int ceil_div(int a, int b) { return (a + b - 1) / b; }

static void launch_gemm(const float* A, const float* B, const float* bias, float* Out,
                        int M, int K, int Ncol, int relu, hipStream_t stream) {
    int waves = (M / 16) * (Ncol / 64);
    int threads = waves * 32;
    int blocks = ceil_div(threads, 256);
    k_gemm_wmma<<<blocks, 256, 0, stream>>>(A, B, bias, Out, M, K, Ncol, relu);
}

extern "C" void kernel_launch(void* const* d_in, const int* in_sizes, int n_in,
                              void* d_out, int out_size, void* d_ws, size_t ws_size,
                              hipStream_t stream) {
    const float* x     = (const float*)d_in[0];
    const int*   ei    = (const int*)  d_in[1];
    const float* w_in  = (const float*)d_in[2];
    const float* b_in  = (const float*)d_in[3];
    const float* w0    = (const float*)d_in[4];
    const float* as0   = (const float*)d_in[5];
    const float* ad0   = (const float*)d_in[6];
    const float* bi0   = (const float*)d_in[7];
    const float* g0    = (const float*)d_in[8];
    const float* be0   = (const float*)d_in[9];
    const float* w1    = (const float*)d_in[10];
    const float* as1   = (const float*)d_in[11];
    const float* ad1   = (const float*)d_in[12];
    const float* bi1   = (const float*)d_in[13];
    const float* g1    = (const float*)d_in[14];
    const float* be1   = (const float*)d_in[15];
    const float* w_out = (const float*)d_in[16];
    const float* b_out = (const float*)d_in[17];
    float* out = (float*)d_out;

    const int N    = in_sizes[0] / 256;   // 50000
    const int E    = in_sizes[1] / 2;     // 800000
    const int Etot = E + N;               // with self-loops

    // Workspace layout (floats)
    float* ws = (float*)d_ws;
    size_t off = 0;
    float* h_in  = ws + off; off += (size_t)N * CH;       // [N,64]
    float* hfeat = ws + off; off += (size_t)N * HC;       // [N,256]
    float* agg   = ws + off; off += (size_t)N * HC;       // [N,256] (also LN output)
    float* eatt  = ws + off; off += (size_t)Etot * HEADS; // [Etot,4]
    float* a_s   = ws + off; off += (size_t)N * HEADS;
    float* a_d   = ws + off; off += (size_t)N * HEADS;
    unsigned int* mkey = (unsigned int*)(ws + off); off += (size_t)N * HEADS;
    float* ssum  = ws + off; off += (size_t)N * HEADS;
    float* meanb = ws + off; off += HC;

    const int nh     = N * HEADS;
    const int eh     = Etot * HEADS;
    const int nhBlk  = ceil_div(nh, 256);
    const int ehBlk  = ceil_div(eh, 256);
    const int aggBlk = ceil_div(Etot, EDGES_PER_BLOCK);
    const int nHcBlk = ceil_div(N * HC, 256);

    // 1) h_in = relu(x @ w_in + b_in)         [N,256]x[256,64]
    launch_gemm(x, w_in, b_in, h_in, N, 256, CH, 1, stream);

    for (int layer = 0; layer < 2; ++layer) {
        const float* W   = layer == 0 ? w0  : w1;
        const float* As  = layer == 0 ? as0 : as1;
        const float* Ad  = layer == 0 ? ad0 : ad1;
        const float* Bi  = layer == 0 ? bi0 : bi1;
        const float* Ga  = layer == 0 ? g0  : g1;
        const float* Be  = layer == 0 ? be0 : be1;
        const float* Ain = layer == 0 ? h_in : agg;   // layer1 input = LN0 output (in agg)
        const int    Kin = layer == 0 ? CH : HC;

        // 2) hfeat = Ain @ W  (bias added after aggregation, per PyG GATConv)
        launch_gemm(Ain, W, nullptr, hfeat, N, Kin, HC, 0, stream);

        // 3) attention coefficients
        k_attn_coef<<<nhBlk, 256, 0, stream>>>(hfeat, As, Ad, a_s, a_d, N);

        // 4) reset segment buffers (agg may alias Ain for layer 1; the GEMM above
        //    has already consumed it — stream order makes this safe)
        k_zero_f32<<<ceil_div(nh, 256), 256, 0, stream>>>((float*)mkey, nh);
        k_zero_f32<<<ceil_div(nh, 256), 256, 0, stream>>>(ssum, nh);
        k_zero_f32<<<nHcBlk, 256, 0, stream>>>(agg, N * HC);

        // 5) segment softmax + weighted aggregation
        k_edge_logits<<<ehBlk, 256, 0, stream>>>(ei, E, Etot, a_s, a_d, eatt, mkey);
        k_edge_exp<<<ehBlk, 256, 0, stream>>>(ei, E, Etot, mkey, eatt, ssum);
        k_edge_aggregate<<<aggBlk, 256, 0, stream>>>(ei, E, Etot, hfeat, eatt, ssum, agg);

        // 6) bias + LayerNorm + ReLU (in place)
        k_bias_ln_relu<<<N, HC, 0, stream>>>(agg, Bi, Ga, Be);
    }

    // 7) mean over nodes, then output projection
    k_zero_f32<<<1, HC, 0, stream>>>(meanb, HC);
    k_colsum<<<256, HC, 0, stream>>>(agg, meanb, N);
    k_final_proj<<<ceil_div(DOUT, 128), 128, 0, stream>>>(meanb, w_out, b_out, out, 1.0f / (float)N);
}